// GRAT3_27642409517703
// MI455X (gfx1250) — compile-verified
//
#include <hip/hip_runtime.h>
#include <hip/hip_bf16.h>
#include <math.h>

// ---------------------------------------------------------------------------
// GRAT (GAT-style) 3-layer pipeline for gfx1250.
// GEMMs use V_WMMA_F32_16X16X4_F32 (exact f32 matrix path on CDNA5).
// Edge passes are L2-resident atomic scatter ops (192MB L2 >> 13MB z-buffer).
// ---------------------------------------------------------------------------

typedef float v2f __attribute__((ext_vector_type(2)));
typedef float v8f __attribute__((ext_vector_type(8)));

#define NEG_SLOPE 0.01f

__device__ __forceinline__ float atomicMaxFloat(float* addr, float value) {
    // Order-preserving int trick; addr initialized to -inf (0xFF800000).
    if (value >= 0.0f)
        return __int_as_float(atomicMax((int*)addr, __float_as_int(value)));
    else
        return __uint_as_float(atomicMin((unsigned int*)addr, __float_as_uint(value)));
}

// ---------------------------------------------------------------------------
// z = (RELU_IN ? relu(A) : A) @ B ;  A: M x K (row major), B: K x 32, z: M x 32
// One wave per 16x16 output tile; K-loop in steps of 4 via f32 WMMA.
// ---------------------------------------------------------------------------
template <int K, bool RELU_IN>
__global__ void gemm_wmma_f32(const float* __restrict__ A,
                              const float* __restrict__ B,
                              float* __restrict__ C, int M) {
    const int wavesPerBlock = blockDim.x >> 5;
    const int wave = blockIdx.x * wavesPerBlock + (threadIdx.x >> 5);
    const int lane = threadIdx.x & 31;
    const int tilesM = (M + 15) >> 4;
    const int tileM = wave >> 1;          // 2 tiles along N (32/16)
    const int tileN = wave & 1;
    if (tileM >= tilesM) return;          // whole-wave uniform exit

    const int halfSel = lane >> 4;        // 0: lanes 0-15, 1: lanes 16-31
    const int l = lane & 15;

    int rowA = tileM * 16 + l;
    if (rowA > M - 1) rowA = M - 1;       // clamp (dup rows, stores guarded)
    const float* __restrict__ arow = A + (size_t)rowA * K;
    const int col = tileN * 16 + l;

    v8f acc = {};
#pragma unroll 4
    for (int k0 = 0; k0 < K; k0 += 4) {
        const int ka = k0 + (halfSel << 1);
        float a0 = arow[ka];
        float a1 = arow[ka + 1];
        if (RELU_IN) { a0 = fmaxf(a0, 0.0f); a1 = fmaxf(a1, 0.0f); }
        v2f af; af.x = a0; af.y = a1;
        v2f bf;
        bf.x = B[(size_t)ka * 32 + col];
        bf.y = B[(size_t)(ka + 1) * 32 + col];
        acc = __builtin_amdgcn_wmma_f32_16x16x4_f32(
            /*neg_a=*/false, af, /*neg_b=*/false, bf,
            /*c_mod=*/(short)0, acc, /*reuse_a=*/false, /*reuse_b=*/false);
    }

    // Store D. Lanes 0-15 hold rows tileM*16+r, lanes 16-31 rows tileM*16+8+r.
    // Fast path: whole 16-row tile in range -> 8 unconditional (clauseable)
    // stores. Only the single ragged boundary tile takes the guarded path.
    float* __restrict__ crow = C + (size_t)(tileM * 16 + halfSel * 8) * 32 + col;
    if (tileM * 16 + 16 <= M) {
#pragma unroll
        for (int r = 0; r < 8; ++r) crow[(size_t)r * 32] = acc[r];
    } else {
        const int rowBase = tileM * 16 + halfSel * 8;
#pragma unroll
        for (int r = 0; r < 8; ++r) {
            if (rowBase + r < M) crow[(size_t)r * 32] = acc[r];
        }
    }
}

// s_src[n] = z[n,:] . a[0:32] ; s_dst[n] = z[n,:] . a[32:64]
__global__ void node_scores(const float* __restrict__ z,
                            const float* __restrict__ a,
                            float* __restrict__ ssrc, float* __restrict__ sdst,
                            int N) {
    const int n = blockIdx.x * blockDim.x + threadIdx.x;
    if (n >= N) return;
    const float* __restrict__ row = z + (size_t)n * 32;
    float s0 = 0.0f, s1 = 0.0f;
#pragma unroll
    for (int k = 0; k < 32; ++k) {
        const float v = row[k];
        s0 = fmaf(v, a[k], s0);
        s1 = fmaf(v, a[32 + k], s1);
    }
    ssrc[n] = s0;
    sdst[n] = s1;
}

// Layer 3 fused: z3 = relu(h2) @ W3 (32x1); s_src = z3*a3[0]; s_dst = z3*a3[1]
__global__ void gemv_fused_l3(const float* __restrict__ h2,
                              const float* __restrict__ W3,
                              const float* __restrict__ a3,
                              float* __restrict__ z3,
                              float* __restrict__ ssrc, float* __restrict__ sdst,
                              int N) {
    const int n = blockIdx.x * blockDim.x + threadIdx.x;
    if (n >= N) return;
    const float* __restrict__ row = h2 + (size_t)n * 32;
    float s = 0.0f;
#pragma unroll
    for (int k = 0; k < 32; ++k) s = fmaf(fmaxf(row[k], 0.0f), W3[k], s);
    z3[n] = s;
    ssrc[n] = s * a3[0];
    sdst[n] = s * a3[1];
}

// m = -inf, denom = 0, out = 0 (grid-stride; re-run every call -> deterministic)
__global__ void init_layer(float* __restrict__ m, float* __restrict__ denom,
                           float* __restrict__ out, int N, int outElems) {
    const int tid = blockIdx.x * blockDim.x + threadIdx.x;
    const int stride = gridDim.x * blockDim.x;
    for (int t = tid; t < N; t += stride) {
        m[t] = -__builtin_inff();
        denom[t] = 0.0f;
    }
    for (int t = tid; t < outElems; t += stride) out[t] = 0.0f;
}

// Pass A: e = leaky_relu(s_src[src] + s_dst[dst]); m[dst] = max(m[dst], e)
__global__ void edge_logits(const int* __restrict__ src, const int* __restrict__ dst,
                            const float* __restrict__ ssrc, const float* __restrict__ sdst,
                            float* __restrict__ e, float* __restrict__ m, int E) {
    const int i = blockIdx.x * blockDim.x + threadIdx.x;
    if (i >= E) return;
    const int s = src[i], d = dst[i];
    float v = ssrc[s] + sdst[d];
    v = (v > 0.0f) ? v : NEG_SLOPE * v;
    e[i] = v;
    atomicMaxFloat(&m[d], v);
}

// Pass B: w = exp(e - m[dst]); denom[dst] += w
__global__ void edge_exp(const int* __restrict__ dst, const float* __restrict__ e,
                         const float* __restrict__ m, float* __restrict__ w,
                         float* __restrict__ denom, int E) {
    const int i = blockIdx.x * blockDim.x + threadIdx.x;
    if (i >= E) return;
    const int d = dst[i];
    const float v = expf(e[i] - m[d]);
    w[i] = v;
    atomicAdd(&denom[d], v);
}

// Pass C (d=32): out[dst,:] += (w/max(denom[dst],1e-9)) * z[src,:]
// One wave (32 lanes) per edge -> fully coalesced row gather/scatter.
__global__ void edge_aggregate32(const int* __restrict__ src, const int* __restrict__ dst,
                                 const float* __restrict__ w, const float* __restrict__ denom,
                                 const float* __restrict__ z, float* __restrict__ out, int E) {
    const int t = blockIdx.x * blockDim.x + threadIdx.x;
    const int edge = t >> 5;
    const int lane = t & 31;
    if (edge >= E) return;
    const int s = src[edge], d = dst[edge];
    const float alpha = w[edge] / fmaxf(denom[d], 1e-9f);
    atomicAdd(&out[(size_t)d * 32 + lane], alpha * z[(size_t)s * 32 + lane]);
}

// Pass C (d=1): out[dst] += alpha * z[src]
__global__ void edge_aggregate1(const int* __restrict__ src, const int* __restrict__ dst,
                                const float* __restrict__ w, const float* __restrict__ denom,
                                const float* __restrict__ z, float* __restrict__ out, int E) {
    const int i = blockIdx.x * blockDim.x + threadIdx.x;
    if (i >= E) return;
    const int s = src[i], d = dst[i];
    const float alpha = w[i] / fmaxf(denom[d], 1e-9f);
    atomicAdd(&out[d], alpha * z[s]);
}

__global__ void sigmoid_inplace(float* __restrict__ out, int N) {
    const int n = blockIdx.x * blockDim.x + threadIdx.x;
    if (n >= N) return;
    out[n] = 1.0f / (1.0f + expf(-out[n]));
}

extern "C" void kernel_launch(void* const* d_in, const int* in_sizes, int n_in,
                              void* d_out, int out_size, void* d_ws, size_t ws_size,
                              hipStream_t stream) {
    const float* feature = (const float*)d_in[0];
    const int*   src     = (const int*)d_in[1];
    const int*   dst     = (const int*)d_in[2];
    const float* W1      = (const float*)d_in[3];
    const float* a1      = (const float*)d_in[4];
    const float* W2      = (const float*)d_in[5];
    const float* a2      = (const float*)d_in[6];
    const float* W3      = (const float*)d_in[7];
    const float* a3      = (const float*)d_in[8];

    const int N = in_sizes[0] / 128;   // 100000
    const int E = in_sizes[1];         // 1600000
    float* out = (float*)d_out;

    // Workspace carve-up (f32): ~53 MB total
    float* ws = (float*)d_ws;
    size_t off = 0;
    float* Z    = ws + off; off += (size_t)N * 32;   // current layer z
    float* H1   = ws + off; off += (size_t)N * 32;   // layer-1 output (pre-relu)
    float* H2   = ws + off; off += (size_t)N * 32;   // layer-2 output (pre-relu)
    float* SS   = ws + off; off += (size_t)N;        // s_src
    float* SD   = ws + off; off += (size_t)N;        // s_dst
    float* MX   = ws + off; off += (size_t)N;        // segment max
    float* DN   = ws + off; off += (size_t)N;        // segment sum (denom)
    float* EB   = ws + off; off += (size_t)E;        // edge logits e
    float* WB   = ws + off; off += (size_t)E;        // edge weights w
    float* Z3   = ws + off; off += (size_t)N;        // layer-3 z
    (void)ws_size; (void)n_in; (void)out_size;

    const int tilesM   = (N + 15) / 16;
    const int gemmWaves = tilesM * 2;                 // 2 N-tiles (32/16)
    const int gemmBlk   = 128;                        // 4 waves/block
    const int gemmGrid  = (gemmWaves + 3) / 4;
    const int nodeGrid  = (N + 255) / 256;
    const int edgeGrid  = (E + 255) / 256;
    const long long aggThreads = (long long)E * 32;
    const int aggGrid   = (int)((aggThreads + 255) / 256);

    // ---------------- Layer 1: 128 -> 32 ----------------
    gemm_wmma_f32<128, false><<<gemmGrid, gemmBlk, 0, stream>>>(feature, W1, Z, N);
    node_scores<<<nodeGrid, 256, 0, stream>>>(Z, a1, SS, SD, N);
    init_layer<<<1024, 256, 0, stream>>>(MX, DN, H1, N, N * 32);
    edge_logits<<<edgeGrid, 256, 0, stream>>>(src, dst, SS, SD, EB, MX, E);
    edge_exp<<<edgeGrid, 256, 0, stream>>>(dst, EB, MX, WB, DN, E);
    edge_aggregate32<<<aggGrid, 256, 0, stream>>>(src, dst, WB, DN, Z, H1, E);

    // ---------------- Layer 2: 32 -> 32 (relu fused into GEMM read) --------
    gemm_wmma_f32<32, true><<<gemmGrid, gemmBlk, 0, stream>>>(H1, W2, Z, N);
    node_scores<<<nodeGrid, 256, 0, stream>>>(Z, a2, SS, SD, N);
    init_layer<<<1024, 256, 0, stream>>>(MX, DN, H2, N, N * 32);
    edge_logits<<<edgeGrid, 256, 0, stream>>>(src, dst, SS, SD, EB, MX, E);
    edge_exp<<<edgeGrid, 256, 0, stream>>>(dst, EB, MX, WB, DN, E);
    edge_aggregate32<<<aggGrid, 256, 0, stream>>>(src, dst, WB, DN, Z, H2, E);

    // ---------------- Layer 3: 32 -> 1 (relu fused) + sigmoid --------------
    gemv_fused_l3<<<nodeGrid, 256, 0, stream>>>(H2, W3, a3, Z3, SS, SD, N);
    init_layer<<<1024, 256, 0, stream>>>(MX, DN, out, N, N);
    edge_logits<<<edgeGrid, 256, 0, stream>>>(src, dst, SS, SD, EB, MX, E);
    edge_exp<<<edgeGrid, 256, 0, stream>>>(dst, EB, MX, WB, DN, E);
    edge_aggregate1<<<edgeGrid, 256, 0, stream>>>(src, dst, WB, DN, Z3, out, E);
    sigmoid_inplace<<<nodeGrid, 256, 0, stream>>>(out, N);
}